// InjectiveMapping_84198538871167
// MI455X (gfx1250) — compile-verified
//
#include <hip/hip_runtime.h>

typedef float v2f __attribute__((ext_vector_type(2)));
typedef float v4f __attribute__((ext_vector_type(4)));
typedef float v8f __attribute__((ext_vector_type(8)));

#define NEG_SLOPE 0.2f
#define NPTS 200000
#define NBAT 16
#define NLAY 10
#define NTILES (NPTS / 16)

// Swap the two 16-lane halves of the wave (group-of-32 swizzle, xor=0x10, and=0x1f).
__device__ __forceinline__ float swapx16(float x) {
    int xi = __builtin_bit_cast(int, x);
    int r  = __builtin_amdgcn_ds_swizzle(xi, 0x401F);
    return __builtin_bit_cast(float, r);
}

__device__ __forceinline__ float lrelu(float x) {
    return x > 0.f ? x : NEG_SLOPE * x;
}

// Streams vertices once; v and the residual stack live in registers across all
// 10 layers. Each 16-point tile is one V_WMMA_F32_16X16X4_F32 per layer:
//   D(16x16) = A(16x4 = M^T, rows 4..15 zero) x B(4x16 = points-as-columns)
// B-rep per lane: lane<16 holds comps (0,1) of point (lane), lane>=16 holds
// comps (2,3) of point (lane-16)  ==> exactly one b64 load/store per lane.
__global__ __launch_bounds__(256) void layers_kernel(
    const float* __restrict__ verts,   // [16, 200000, 4]
    const float* __restrict__ wts,     // [10, 16, 20]
    float* __restrict__ vout) {        // [16, 200000, 4]
    const int b    = blockIdx.y;
    const int lane = threadIdx.x & 31;
    const int hi   = lane >> 4;     // 0: lower half (K=0,1), 1: upper half (K=2,3)
    const int c    = lane & 15;     // column within half (point index / A row)

    const int wave   = blockIdx.x * (blockDim.x >> 5) + (threadIdx.x >> 5);
    const int nwaves = gridDim.x * (blockDim.x >> 5);

    // Preload per-layer A = M^T (2 VGPRs) and bias in B-rep (2 VGPRs).
    v2f   A[NLAY];
    float bb0[NLAY], bb1[NLAY];
#pragma unroll
    for (int l = 0; l < NLAY; ++l) {
        const float* wl = wts + (l * NBAT + b) * 20;
        const int d0 = hi ? 2 : 0;           // K row pair handled by this half
        float a0 = 0.f, a1 = 0.f;
        if (c < 4) {                         // A[m=c, k=d] = M[d, c] = wl[d*4+c]
            a0 = wl[d0 * 4 + c];
            a1 = wl[(d0 + 1) * 4 + c];
        }
        A[l].x = a0;
        A[l].y = a1;
        bb0[l] = wl[16 + d0];                // bias for comp d0   (this half)
        bb1[l] = wl[16 + d0 + 1];            // bias for comp d0+1 (this half)
    }

    const float* vin = verts + (size_t)b * NPTS * 4;
    float*       vo  = vout  + (size_t)b * NPTS * 4;
    const int laneoff = c * 4 + hi * 2;      // float offset within a 16-pt tile

    for (int t = wave; t < NTILES; t += nwaves) {
        const int base = t * 64 + laneoff;
        v2f bv = __builtin_nontemporal_load((const v2f*)(vin + base));
        float B0 = bv.x, B1 = bv.y;
        float r00 = 0.f, r01 = 0.f, r10 = 0.f, r11 = 0.f;
        float r20 = 0.f, r21 = 0.f, r30 = 0.f, r31 = 0.f;

#pragma unroll
        for (int l = 0; l < NLAY; ++l) {
            v2f a = A[l];
            v2f bm;
            bm.x = B0;
            bm.y = B1;
            v8f cz = {0.f, 0.f, 0.f, 0.f, 0.f, 0.f, 0.f, 0.f};
            // (neg_a, A, neg_b, B, c_mod, C, reuse_a, reuse_b)
            v8f d = __builtin_amdgcn_wmma_f32_16x16x4_f32(
                false, a, false, bm, (short)0, cz, false, false);

            // D rows e=0..3 sit in d[0..3] lanes 0-15.  Rebuild B-rep:
            // swizzles must run with full EXEC, then per-lane select.
            float d2s = swapx16(d[2]);
            float d3s = swapx16(d[3]);
            float t0 = hi ? d2s : d[0];
            float t1 = hi ? d3s : d[1];

            t0 = lrelu(t0 + bb0[l]);
            t1 = lrelu(t1 + bb1[l]);

            // LIFO residual skips: push at 0..3, pop+add at 6..9.
            if (l == 0)      { r00 = t0; r01 = t1; }
            else if (l == 1) { r10 = t0; r11 = t1; }
            else if (l == 2) { r20 = t0; r21 = t1; }
            else if (l == 3) { r30 = t0; r31 = t1; }
            else if (l == 6) { t0 += r30; t1 += r31; }
            else if (l == 7) { t0 += r20; t1 += r21; }
            else if (l == 8) { t0 += r10; t1 += r11; }
            else if (l == 9) { t0 += r00; t1 += r01; }

            B0 = t0;
            B1 = t1;
        }

        v2f ov;
        ov.x = B0;
        ov.y = B1;
        __builtin_nontemporal_store(ov, (v2f*)(vo + base));
    }
}

// det_rep: 160 scalar 4x4 determinants broadcast over 200000 points each
// (128 MB pure fill) -- NT b128 stores, det recomputed per block (free).
__global__ __launch_bounds__(256) void det_fill_kernel(
    const float* __restrict__ wts,    // [10, 16, 20]
    float* __restrict__ dout) {       // [10, 16, 200000]
    const int lb = blockIdx.y;        // l*16 + b
    const float* m = wts + lb * 20;
    const float m00 = m[0],  m01 = m[1],  m02 = m[2],  m03 = m[3];
    const float m10 = m[4],  m11 = m[5],  m12 = m[6],  m13 = m[7];
    const float m20 = m[8],  m21 = m[9],  m22 = m[10], m23 = m[11];
    const float m30 = m[12], m31 = m[13], m32 = m[14], m33 = m[15];

    const float s0 = m22 * m33 - m23 * m32;
    const float s1 = m21 * m33 - m23 * m31;
    const float s2 = m21 * m32 - m22 * m31;
    const float s3 = m20 * m33 - m23 * m30;
    const float s4 = m20 * m32 - m22 * m30;
    const float s5 = m20 * m31 - m21 * m30;

    const float det =
          m00 * (m10 * 0.f + m11 * s0 - m12 * s1 + m13 * s2)
        - m01 * (m10 * s0 - m12 * s3 + m13 * s4)
        + m02 * (m10 * s1 - m11 * s3 + m13 * s5)
        - m03 * (m10 * s2 - m11 * s4 + m12 * s5);

    v4f dv = {det, det, det, det};
    float* o = dout + (size_t)lb * NPTS;
    const int stride = gridDim.x * blockDim.x;
    for (int i = blockIdx.x * blockDim.x + threadIdx.x; i < NPTS / 4; i += stride) {
        __builtin_nontemporal_store(dv, (v4f*)(o + i * 4));
    }
}

extern "C" void kernel_launch(void* const* d_in, const int* in_sizes, int n_in,
                              void* d_out, int out_size, void* d_ws, size_t ws_size,
                              hipStream_t stream) {
    const float* verts = (const float*)d_in[0];   // [16, 200000, 4]
    const float* wts   = (const float*)d_in[1];   // [10, 16, 20]
    float* out  = (float*)d_out;
    float* vout = out;                                  // 16*200000*4 floats
    float* dout = out + (size_t)NBAT * NPTS * 4;        // 10*16*200000 floats

    dim3 g1(104, NBAT);            // 832 waves/batch, ~15 tiles each
    layers_kernel<<<g1, 256, 0, stream>>>(verts, wts, vout);

    dim3 g2(64, NLAY * NBAT);      // 160 (l,b) planes, 50000 b128 stores each
    det_fill_kernel<<<g2, 256, 0, stream>>>(wts, dout);
}